// Learnable_MMF_43576738185283
// MI455X (gfx1250) — compile-verified
//
#include <hip/hip_runtime.h>

#define NDIM 2048
#define LLEV 512
#define KROT 16

typedef __attribute__((ext_vector_type(2))) float v2f;
typedef __attribute__((ext_vector_type(8))) float v8f;

// Async global -> LDS DMA (CDNA5, tracked by ASYNCcnt). Each enabled lane
// copies 16B (b128) or 4B (b32) from its global address to its LDS address.
__device__ __forceinline__ void async_copy_b128(const void* gptr, unsigned lds_off) {
  asm volatile("global_load_async_to_lds_b128 %0, %1, off"
               :: "v"(lds_off), "v"(gptr) : "memory");
}
__device__ __forceinline__ void async_copy_b32(const void* gptr, unsigned lds_off) {
  asm volatile("global_load_async_to_lds_b32 %0, %1, off"
               :: "v"(lds_off), "v"(gptr) : "memory");
}
__device__ __forceinline__ void wait_async0() {
  asm volatile("s_wait_asynccnt 0x0" ::: "memory");
}
// LDS aperture: low 32 bits of a generic LDS pointer are the LDS byte offset.
__device__ __forceinline__ unsigned lds_addr(const void* p) {
  return (unsigned)(uintptr_t)p;
}

// ---------------------------------------------------------------------------
// Copy A -> working D region (scan mutates it; inputs must stay intact)
// ---------------------------------------------------------------------------
__global__ void copy_kernel(const float* __restrict__ src, float* __restrict__ dst) {
  int i = blockIdx.x * blockDim.x + threadIdx.x;
  dst[i] = src[i];
}

// right0 = I
__global__ void init_eye_kernel(float* __restrict__ R) {
  int i = blockIdx.x * blockDim.x + threadIdx.x;
  int r = i >> 11, c = i & (NDIM - 1);
  R[i] = (r == c) ? 1.0f : 0.0f;
}

// ---------------------------------------------------------------------------
// Scan level: row updates  A[il,:] = O @ A[il,:]  and  right[il,:] = O @ right[il,:]
// One thread per column of A or right (gid < NDIM -> A, else right).
// Safe in-place: each thread reads all 16 source rows before writing.
// ---------------------------------------------------------------------------
__global__ void scan_rows_kernel(float* __restrict__ Aw, float* __restrict__ Rw,
                                 const float* __restrict__ O,
                                 const int* __restrict__ idx) {
  __shared__ float sO[KROT * KROT];
  __shared__ int sIdx[KROT];
  int t = threadIdx.x;
  sO[t] = O[t];
  if (t < KROT) sIdx[t] = idx[t];
  __syncthreads();

  int gid = blockIdx.x * 256 + t;                 // 0 .. 4095
  float* M = (gid < NDIM) ? Aw : Rw;
  int c = gid & (NDIM - 1);

  float v[KROT];
#pragma unroll
  for (int k = 0; k < KROT; ++k) v[k] = M[(size_t)sIdx[k] * NDIM + c];
#pragma unroll
  for (int j = 0; j < KROT; ++j) {
    float acc = 0.0f;
#pragma unroll
    for (int k = 0; k < KROT; ++k) acc = fmaf(sO[j * KROT + k], v[k], acc);
    M[(size_t)sIdx[j] * NDIM + c] = acc;
  }
}

// ---------------------------------------------------------------------------
// Scan level: column update  A[:,il] = A[:,il] @ O^T
//   out[r][j] = sum_k A[r, il[k]] * O[j][k]
// One thread per row; runs AFTER scan_rows (reads post-row-update values).
// ---------------------------------------------------------------------------
__global__ void scan_cols_kernel(float* __restrict__ Aw,
                                 const float* __restrict__ O,
                                 const int* __restrict__ idx) {
  __shared__ float sO[KROT * KROT];
  __shared__ int sIdx[KROT];
  int t = threadIdx.x;
  sO[t] = O[t];
  if (t < KROT) sIdx[t] = idx[t];
  __syncthreads();

  int r = blockIdx.x * 256 + t;
  size_t base = (size_t)r * NDIM;
  float w[KROT];
#pragma unroll
  for (int k = 0; k < KROT; ++k) w[k] = Aw[base + sIdx[k]];
#pragma unroll
  for (int j = 0; j < KROT; ++j) {
    float acc = 0.0f;
#pragma unroll
    for (int k = 0; k < KROT; ++k) acc = fmaf(w[k], sO[j * KROT + k], acc);
    Aw[base + sIdx[j]] = acc;
  }
}

// ---------------------------------------------------------------------------
// Build active flags (single block, ordered by __syncthreads)
// ---------------------------------------------------------------------------
__global__ void flags_kernel(int* __restrict__ flags,
                             const int* __restrict__ active_idx) {
  int t = threadIdx.x;                 // 1024 threads
  flags[t] = 0;
  flags[t + 1024] = 0;
  __syncthreads();
  flags[active_idx[t]] = 1;
}

// D = A_out * left ; left diag == 1, off-diag == active_i * active_j
__global__ void mask_kernel(float* __restrict__ D, const int* __restrict__ flags) {
  int i = blockIdx.x * blockDim.x + threadIdx.x;
  int r = i >> 11, c = i & (NDIM - 1);
  float v = D[i];
  if (r != c && !(flags[r] & flags[c])) v = 0.0f;
  D[i] = v;
}

// ---------------------------------------------------------------------------
// f32 WMMA GEMM:  C = op(X) @ Y   (all NDIM x NDIM, row-major)
//   TRANS_A = false : op(X) = X
//   TRANS_A = true  : op(X) = X^T   (used for right^T @ T)
// 128x128 block tile, BK=32, 256 threads = 8 waves (2x4), wave tile 64x32.
// Tiles staged with GLOBAL_LOAD_ASYNC_TO_LDS (ASYNCcnt), compute with
// V_WMMA_F32_16X16X4_F32 (D = A(16x4) * B(4x16) + C).
// LDS pads: Xs stride 36 floats (144B, 16B-aligned chunks; 36*m mod 64 banks
// distinct for m=0..15), Ys stride 132 floats (528B, 16B-aligned chunks).
// ---------------------------------------------------------------------------
template <bool TRANS_A>
__global__ __launch_bounds__(256) void gemm_wmma_kernel(
    const float* __restrict__ X, const float* __restrict__ Y,
    float* __restrict__ C) {
  constexpr int BM = 128, BN = 128, BK = 32;
  constexpr int XPAD = 4, YPAD = 4;
  __shared__ alignas(16) float Xs[BM][BK + XPAD];   // 18.0 KB
  __shared__ alignas(16) float Ys[BK][BN + YPAD];   // 16.5 KB

  const int tid  = threadIdx.x;
  const int lane = tid & 31;
  const int wave = tid >> 5;     // 0..7
  const int wm   = wave >> 2;    // 0..1  -> 64 rows each
  const int wn   = wave & 3;     // 0..3  -> 32 cols each
  const int bm   = blockIdx.y * BM;
  const int bn   = blockIdx.x * BN;

  const int frag = lane & 15;            // M (A-frag) or N (B-frag)
  const int klo  = (lane >> 4) << 1;     // lanes 16-31 hold K+2 / K+3
  const int mhi  = (lane >> 4) << 3;     // C/D: lanes 16-31 hold M+8

  v8f acc[4][2];
#pragma unroll
  for (int sm = 0; sm < 4; ++sm)
#pragma unroll
    for (int sn = 0; sn < 2; ++sn) acc[sm][sn] = {};

  for (int kt = 0; kt < NDIM; kt += BK) {
    // ---- async-stage X tile (128 x 32) ----
    if (TRANS_A) {
      // X^T: global is strided along k -> per-element async B32
#pragma unroll
      for (int i = 0; i < 16; ++i) {
        int e = tid + i * 256;          // 0..4095
        int m = e >> 5, k = e & 31;
        async_copy_b32(X + (size_t)(kt + k) * NDIM + (bm + m),
                       lds_addr(&Xs[m][k]));
      }
    } else {
      // 1024 16-byte chunks (8 per row), 4 per thread
#pragma unroll
      for (int i = 0; i < 4; ++i) {
        int cid = tid + i * 256;        // 0..1023
        int m = cid >> 3, kq = (cid & 7) << 2;
        async_copy_b128(X + (size_t)(bm + m) * NDIM + (kt + kq),
                        lds_addr(&Xs[m][kq]));
      }
    }
    // ---- async-stage Y tile (32 x 128): 1024 chunks, 4 per thread ----
#pragma unroll
    for (int i = 0; i < 4; ++i) {
      int cid = tid + i * 256;          // 0..1023
      int k = cid >> 5, nq = (cid & 31) << 2;
      async_copy_b128(Y + (size_t)(kt + k) * NDIM + (bn + nq),
                      lds_addr(&Ys[k][nq]));
    }
    wait_async0();        // this wave's DMA done
    __syncthreads();      // all waves' DMA visible

#pragma unroll
    for (int kk = 0; kk < BK; kk += 4) {
      v2f a[4], b[2];
#pragma unroll
      for (int sm = 0; sm < 4; ++sm) {
        int m = wm * 64 + sm * 16 + frag;
        a[sm].x = Xs[m][kk + klo];
        a[sm].y = Xs[m][kk + klo + 1];
      }
#pragma unroll
      for (int sn = 0; sn < 2; ++sn) {
        int n = wn * 32 + sn * 16 + frag;
        b[sn].x = Ys[kk + klo][n];
        b[sn].y = Ys[kk + klo + 1][n];
      }
#pragma unroll
      for (int sm = 0; sm < 4; ++sm)
#pragma unroll
        for (int sn = 0; sn < 2; ++sn)
          acc[sm][sn] = __builtin_amdgcn_wmma_f32_16x16x4_f32(
              false, a[sm], false, b[sn], (short)0, acc[sm][sn], false, false);
    }
    __syncthreads();      // done reading before next stage overwrites
  }

  // ---- epilogue ----
#pragma unroll
  for (int sm = 0; sm < 4; ++sm)
#pragma unroll
    for (int sn = 0; sn < 2; ++sn)
#pragma unroll
      for (int v = 0; v < 8; ++v) {
        int row = bm + wm * 64 + sm * 16 + v + mhi;
        int col = bn + wn * 32 + sn * 16 + frag;
        C[(size_t)row * NDIM + col] = acc[sm][sn][v];
      }
}

// ---------------------------------------------------------------------------
// Gathers
// ---------------------------------------------------------------------------
__global__ void gather_coeff_kernel(const float* __restrict__ D,
                                    const int* __restrict__ ri,
                                    const int* __restrict__ ci,
                                    float* __restrict__ out) {
  int i = blockIdx.x * blockDim.x + threadIdx.x;   // 1024*1024
  int r = i >> 10, c = i & 1023;
  out[i] = D[(size_t)ri[r] * NDIM + ci[c]];
}

__global__ void gather_rows_kernel(const float* __restrict__ R,
                                   const int* __restrict__ ri,
                                   float* __restrict__ out) {
  int i = blockIdx.x * blockDim.x + threadIdx.x;   // 1024*2048
  int r = i >> 11, c = i & (NDIM - 1);
  out[i] = R[(size_t)ri[r] * NDIM + c];
}

// ---------------------------------------------------------------------------
extern "C" void kernel_launch(void* const* d_in, const int* in_sizes, int n_in,
                              void* d_out, int out_size, void* d_ws, size_t ws_size,
                              hipStream_t stream) {
  const float* A          = (const float*)d_in[0];
  const float* O          = (const float*)d_in[1];
  const int*   idx        = (const int*)d_in[2];
  const int*   active_idx = (const int*)d_in[3];
  const int*   inact_idx  = (const int*)d_in[4];

  float* out = (float*)d_out;
  const size_t NN = (size_t)NDIM * NDIM;           // 4,194,304
  float* Arec = out + 0;                           // 2048x2048
  float* Rw   = out + NN;                          // right (working + output)
  float* Dw   = out + 2 * NN;                      // A working copy -> D
  float* MC   = out + 3 * NN;                      // 1024x1024
  float* FC   = MC + 1024 * 1024;                  // 1024x1024
  float* MW   = FC + 1024 * 1024;                  // 1024x2048
  float* FW   = MW + 1024 * 2048;                  // 1024x2048
  float* T    = MC;                                // 16MB scratch aliases MC/FC/MW
  int*   flags = (int*)d_ws;                       // 2048 ints

  // init working copies
  copy_kernel<<<NN / 256, 256, 0, stream>>>(A, Dw);
  init_eye_kernel<<<NN / 256, 256, 0, stream>>>(Rw);

  // sequential scan over 512 levels (serially dependent)
  for (int l = 0; l < LLEV; ++l) {
    scan_rows_kernel<<<16, 256, 0, stream>>>(Dw, Rw, O + l * KROT * KROT,
                                             idx + l * KROT);
    scan_cols_kernel<<<8, 256, 0, stream>>>(Dw, O + l * KROT * KROT,
                                            idx + l * KROT);
  }

  // mask A_out -> D (in place)
  flags_kernel<<<1, 1024, 0, stream>>>(flags, active_idx);
  mask_kernel<<<NN / 256, 256, 0, stream>>>(Dw, flags);

  // A_rec = right^T @ D @ right   (two f32 WMMA GEMMs)
  dim3 ggrid(NDIM / 128, NDIM / 128);
  gemm_wmma_kernel<false><<<ggrid, 256, 0, stream>>>(Dw, Rw, T);    // T = D @ right
  gemm_wmma_kernel<true><<<ggrid, 256, 0, stream>>>(Rw, T, Arec);   // A_rec = right^T @ T

  // gathers (overwrite the T scratch region after GEMM2 is done)
  gather_coeff_kernel<<<(1024 * 1024) / 256, 256, 0, stream>>>(Dw, inact_idx, inact_idx, MC);
  gather_coeff_kernel<<<(1024 * 1024) / 256, 256, 0, stream>>>(Dw, active_idx, active_idx, FC);
  gather_rows_kernel<<<(1024 * 2048) / 256, 256, 0, stream>>>(Rw, inact_idx, MW);
  gather_rows_kernel<<<(1024 * 2048) / 256, 256, 0, stream>>>(Rw, active_idx, FW);
}